// AspireConSent_26018911879771
// MI455X (gfx1250) — compile-verified
//
#include <hip/hip_runtime.h>

typedef __attribute__((ext_vector_type(2))) float v2f;
typedef __attribute__((ext_vector_type(8))) float v8f;

#define B_  64
#define S_  512
#define D_  768
#define MS  20   // max_sents (output rows kept)

// ---------------------------------------------------------------------------
// Kernel 1: per-batch histogram of sent_ids -> reciprocal counts in workspace,
//           plus the doc_cls copy (hidden[:,0,:] -> out[0 : B*D]).
// ---------------------------------------------------------------------------
__global__ void __launch_bounds__(256)
seg_prep_kernel(const float* __restrict__ hidden,
                const int*   __restrict__ sent_ids,
                float*       __restrict__ out_doc,
                float*       __restrict__ inv) {
    const int b   = blockIdx.x;
    const int tid = threadIdx.x;

    __shared__ int cnt[32];
    if (tid < 32) cnt[tid] = 0;
    __syncthreads();

    const int* sid = sent_ids + b * S_;
    for (int s = tid; s < S_; s += 256) {
        atomicAdd(&cnt[sid[s]], 1);           // sid in [0, 20]
    }
    __syncthreads();

    if (tid < MS) {
        float c = (float)cnt[tid];
        inv[b * 32 + tid] = 1.0f / fmaxf(c, 1.0f);
    }

    // doc_cls copy: hidden[b, 0, :]
    const float* src = hidden + (size_t)b * S_ * D_;
    float*       dst = out_doc + (size_t)b * D_;
    for (int d = tid; d < D_; d += 256) dst[d] = src[d];
}

// ---------------------------------------------------------------------------
// Kernel 2: segment-sum as one-hot matmul using V_WMMA_F32_16X16X4_F32.
// Grid: (B, 12) blocks of 128 threads (4 waves). Wave w of block (b, gy)
// owns batch b, column tile n0 = (gy*4 + w)*16, streams K = 0..511 step 4.
// Two WMMAs (M-tiles 0-15 and 16-31 of the one-hot matrix) share each B load,
// so `hidden` is read exactly once -> HBM-bandwidth bound (~100 MB total).
// ---------------------------------------------------------------------------
__global__ void __launch_bounds__(128)
seg_wmma_kernel(const float* __restrict__ hidden,
                const int*   __restrict__ sent_ids,
                const float* __restrict__ inv,
                float*       __restrict__ out_sent) {
    const int b     = blockIdx.x;
    const int wave  = threadIdx.x >> 5;
    const int lane  = threadIdx.x & 31;
    const int ntile = blockIdx.y * 4 + wave;   // 0..47
    const int n0    = ntile * 16;
    const int hi    = lane >> 4;               // lane half selects K pair
    const int ml    = lane & 15;               // M (for A) / N (for B/D)
    const int col   = n0 + ml;

    const float* hb  = hidden   + (size_t)b * S_ * D_;
    const int*   sid = sent_ids + b * S_;

    v8f acc0 = {};   // rows  0..15 of one-hot P
    v8f acc1 = {};   // rows 16..31 (only 16..19 stored)

    #pragma unroll 4
    for (int k0 = 0; k0 < S_; k0 += 4) {
        const int kk = k0 + 2 * hi;

        // B-tile (4x16 f32): VGPR0 = row kk, VGPR1 = row kk+1, lane = column.
        v2f bm;
        bm.x = hb[(size_t)kk * D_ + col];
        bm.y = hb[(size_t)(kk + 1) * D_ + col];

        // Prefetch the rows 8 ahead (speculative; OOB prefetch is dropped).
        __builtin_prefetch(&hb[(size_t)(kk + 8) * D_ + col], 0, 1);

        // A-tiles (16x4 one-hot): lane M = ml, VGPR0/1 = K = kk, kk+1.
        const int s0 = sid[kk];
        const int s1 = sid[kk + 1];
        v2f a0, a1;
        a0.x = (s0 == ml)      ? 1.0f : 0.0f;
        a0.y = (s1 == ml)      ? 1.0f : 0.0f;
        a1.x = (s0 == ml + 16) ? 1.0f : 0.0f;
        a1.y = (s1 == ml + 16) ? 1.0f : 0.0f;

        acc0 = __builtin_amdgcn_wmma_f32_16x16x4_f32(
            /*neg_a=*/false, a0, /*neg_b=*/false, bm,
            /*c_mod=*/(short)0, acc0, /*reuse_a=*/false, /*reuse_b=*/false);
        acc1 = __builtin_amdgcn_wmma_f32_16x16x4_f32(
            /*neg_a=*/false, a1, /*neg_b=*/false, bm,
            /*c_mod=*/(short)0, acc1, /*reuse_a=*/false, /*reuse_b=*/false);
    }

    // D layout: VGPR r, lanes 0-15 -> M=r, lanes 16-31 -> M=r+8, N=ml.
    const float* invb = inv + b * 32;

    #pragma unroll
    for (int r = 0; r < 8; ++r) {
        const int m = r + 8 * hi;              // 0..15, all < MS
        out_sent[((size_t)(b * MS + m)) * D_ + col] = acc0[r] * invb[m];
    }
    #pragma unroll
    for (int r = 0; r < 8; ++r) {
        const int m = 16 + r + 8 * hi;         // only 16..19 kept
        if (m < MS)
            out_sent[((size_t)(b * MS + m)) * D_ + col] = acc1[r] * invb[m];
    }
}

// ---------------------------------------------------------------------------
extern "C" void kernel_launch(void* const* d_in, const int* in_sizes, int n_in,
                              void* d_out, int out_size, void* d_ws, size_t ws_size,
                              hipStream_t stream) {
    const float* hidden   = (const float*)d_in[0];
    const int*   sent_ids = (const int*)d_in[1];
    // d_in[2] = max_sents scalar (== 20), hardcoded as MS.

    float* out      = (float*)d_out;
    float* inv      = (float*)d_ws;            // B*32 floats of scratch
    float* out_sent = out + (size_t)B_ * D_;   // after doc_cls section

    seg_prep_kernel<<<dim3(B_), dim3(256), 0, stream>>>(hidden, sent_ids, out, inv);
    seg_wmma_kernel<<<dim3(B_, 12), dim3(128), 0, stream>>>(hidden, sent_ids, inv, out_sent);
}